// GE2ELoss_43319040147609
// MI455X (gfx1250) — compile-verified
//
#include <hip/hip_runtime.h>
#include <math.h>

typedef __attribute__((ext_vector_type(2))) float v2f;
typedef __attribute__((ext_vector_type(8))) float v8f;

#define EPS 1e-8f
#define N_SPK 512
#define U_UTT 32
#define D_DIM 512
#define NROWS (N_SPK * U_UTT)   // 16384
#define MROWS 32                // rows per block (2 M-tiles, B reused 2x)
#define LDSTRIDE 516            // 512 + 4 pad: conflict-free ds_load_b64 A-frags

// ---------------------------------------------------------------------------
// Kernel 1: centroids = mean over utterances, normalized, stored row-pair
// interleaved:  cP[(d>>1)*1024 + j*2 + (d&1)] = c_n[j][d]
// -> matmul B fragment (rows K,K+1 at column j) is one contiguous b64 load.
// ---------------------------------------------------------------------------
__global__ __launch_bounds__(256) void k_centroid(const float* __restrict__ e,
                                                  float* __restrict__ cP) {
    __shared__ float red[256];
    const int j   = blockIdx.x;     // speaker
    const int tid = threadIdx.x;
    const float* base = e + (size_t)j * U_UTT * D_DIM;
    float c0 = 0.f, c1 = 0.f;
    #pragma unroll 4
    for (int k = 0; k < U_UTT; ++k) {
        c0 += base[k * D_DIM + tid];
        c1 += base[k * D_DIM + tid + 256];
    }
    c0 *= (1.0f / U_UTT);
    c1 *= (1.0f / U_UTT);
    red[tid] = c0 * c0 + c1 * c1;
    __syncthreads();
    for (int s = 128; s > 0; s >>= 1) {
        if (tid < s) red[tid] += red[tid + s];
        __syncthreads();
    }
    const float inv = 1.0f / fmaxf(sqrtf(red[0]), EPS);
    const int d0 = tid, d1 = tid + 256;
    cP[(size_t)(d0 >> 1) * 1024 + j * 2 + (d0 & 1)] = c0 * inv;
    cP[(size_t)(d1 >> 1) * 1024 + j * 2 + (d1 & 1)] = c1 * inv;
}

// ---------------------------------------------------------------------------
// Kernel 2: per-row inverse norms, one wave32 per row
// ---------------------------------------------------------------------------
__global__ __launch_bounds__(128) void k_rownorm(const float* __restrict__ e,
                                                 float* __restrict__ invn) {
    const int warp = threadIdx.x >> 5;
    const int lane = threadIdx.x & 31;
    const int row  = blockIdx.x * 4 + warp;
    const float* p = e + (size_t)row * D_DIM;
    float s = 0.f;
    #pragma unroll 4
    for (int i = lane; i < D_DIM; i += 32) {
        float x = p[i];
        s += x * x;
    }
    #pragma unroll
    for (int off = 16; off > 0; off >>= 1) s += __shfl_xor(s, off, 32);
    if (lane == 0) invn[row] = 1.0f / fmaxf(sqrtf(s), EPS);
}

// ---------------------------------------------------------------------------
// Kernel 3: WMMA matmul (32 rows x 512 centroids per block) + fused softmax
// ---------------------------------------------------------------------------
__global__ __launch_bounds__(128) void k_ge2e(const float* __restrict__ e,
                                              const float* __restrict__ cP,
                                              const float* __restrict__ invn,
                                              const float* __restrict__ wp,
                                              const float* __restrict__ bp,
                                              float* __restrict__ partials) {
    __shared__ float lds[MROWS * LDSTRIDE + MROWS]; // phase1: A (32x512, stride 516)
                                                    // phase2: logits (same layout)
                                                    // tail: 32 row partials
    const int tid  = threadIdx.x;
    const int wave = tid >> 5;
    const int lane = tid & 31;
    const int r0   = blockIdx.x * MROWS;
    const float wv = wp[0];
    const float bv = bp[0];

    // ---- stage A tile (32 x 512) into LDS (float4, coalesced) ----
    {
        const float4* src = (const float4*)(e + (size_t)r0 * D_DIM);
        for (int i = tid; i < MROWS * D_DIM / 4; i += 128) {
            float4 v = src[i];
            int row = (i * 4) / D_DIM;
            int col = (i * 4) % D_DIM;
            *(float4*)(&lds[row * LDSTRIDE + col]) = v;
        }
    }
    __syncthreads();

    const int half = lane >> 4;          // 0: lanes 0-15, 1: lanes 16-31
    const int l16  = lane & 15;
    const int koff = half * 2;           // A/B frag K offset per ISA layout
    const int n0   = wave * 128;         // this wave's 128 columns (8 N-tiles)

    v8f acc0[8], acc1[8];                // M-tile 0 (rows 0-15), M-tile 1 (16-31)
    const v8f zero = {0.f, 0.f, 0.f, 0.f, 0.f, 0.f, 0.f, 0.f};
    #pragma unroll
    for (int q = 0; q < 8; ++q) { acc0[q] = zero; acc1[q] = zero; }

    // B base: pair index half, column n0 + l16 -> one b64 = rows (K,K+1)
    const float* bb = cP + (size_t)half * 1024 + (size_t)(n0 + l16) * 2;

    for (int k = 0; k < D_DIM; k += 4) {
        // A fragments: lanes 0-15 -> {K=k,k+1}, lanes 16-31 -> {k+2,k+3}
        v2f a0 = *(const v2f*)(&lds[l16 * LDSTRIDE + k + koff]);
        v2f a1 = *(const v2f*)(&lds[(l16 + 16) * LDSTRIDE + k + koff]);
        const float* bk = bb + (size_t)k * 512;   // (k/2)*1024
        #pragma unroll
        for (int q = 0; q < 8; ++q) {
            v2f b = *(const v2f*)(bk + q * 32);   // global_load_b64
            acc0[q] = __builtin_amdgcn_wmma_f32_16x16x4_f32(
                false, a0, false, b, (short)0, acc0[q], false, false);
            acc1[q] = __builtin_amdgcn_wmma_f32_16x16x4_f32(
                false, a1, false, b, (short)0, acc1[q], false, false);
        }
    }
    __syncthreads();   // all waves done reading A -> LDS reusable

    // ---- scale (w * dot * 1/||row|| + b) and write logits into LDS ----
    // C/D layout: VGPR v, lanes 0-15 -> M=v ; lanes 16-31 -> M=v+8
    const float* ivp0 = invn + r0 + 8 * half;
    const float* ivp1 = ivp0 + 16;
    float sc0[8], sc1[8];
    #pragma unroll
    for (int v = 0; v < 8; ++v) { sc0[v] = ivp0[v] * wv; sc1[v] = ivp1[v] * wv; }
    #pragma unroll
    for (int q = 0; q < 8; ++q) {
        #pragma unroll
        for (int v = 0; v < 8; ++v) {
            int row = v + 8 * half;
            lds[row * LDSTRIDE + n0 + q * 16 + l16]        = acc0[q][v] * sc0[v] + bv;
            lds[(row + 16) * LDSTRIDE + n0 + q * 16 + l16] = acc1[q][v] * sc1[v] + bv;
        }
    }
    __syncthreads();

    // ---- fused log-softmax: 4 threads per row, online (max, sumexp) ----
    const int row = tid >> 2;            // 0..31
    const int sub = tid & 3;
    const float* srow = &lds[row * LDSTRIDE];
    float m = -INFINITY, s = 0.f;
    for (int j = sub; j < N_SPK; j += 4) {
        float x  = srow[j];
        float nm = fmaxf(m, x);
        s = s * __expf(m - nm) + __expf(x - nm);
        m = nm;
    }
    #pragma unroll
    for (int off = 1; off < 4; off <<= 1) {
        float mo = __shfl_xor(m, off, 32);
        float so = __shfl_xor(s, off, 32);
        float nm = fmaxf(m, mo);
        s = s * __expf(m - nm) + so * __expf(mo - nm);
        m = nm;
    }
    const int jdiag = blockIdx.x;        // r0/32: all 32 rows share one speaker
    if (sub == 0) {
        float logZ = m + __logf(s);
        lds[MROWS * LDSTRIDE + row] = logZ - srow[jdiag];   // -logp_diag
    }
    __syncthreads();
    if (tid == 0) {
        float t = 0.f;
        #pragma unroll
        for (int r = 0; r < MROWS; ++r) t += lds[MROWS * LDSTRIDE + r];
        partials[blockIdx.x] = t;
    }
}

// ---------------------------------------------------------------------------
// Kernel 4: final reduction of 512 block partials -> scalar loss
// ---------------------------------------------------------------------------
__global__ __launch_bounds__(256) void k_final(const float* __restrict__ partials,
                                               float* __restrict__ out) {
    __shared__ float red[256];
    const int tid = threadIdx.x;
    float s = 0.f;
    for (int i = tid; i < NROWS / MROWS; i += 256) s += partials[i];
    red[tid] = s;
    __syncthreads();
    for (int st = 128; st > 0; st >>= 1) {
        if (tid < st) red[tid] += red[tid + st];
        __syncthreads();
    }
    if (tid == 0) out[0] = red[0] / (float)NROWS;
}

// ---------------------------------------------------------------------------
extern "C" void kernel_launch(void* const* d_in, const int* in_sizes, int n_in,
                              void* d_out, int out_size, void* d_ws, size_t ws_size,
                              hipStream_t stream) {
    const float* e = (const float*)d_in[0];   // (512, 32, 512) f32
    const float* w = (const float*)d_in[1];   // scalar
    const float* b = (const float*)d_in[2];   // scalar

    float* ws       = (float*)d_ws;
    float* cP       = ws;                         // 512*512 floats (1 MB), packed
    float* invn     = cP + (size_t)D_DIM * N_SPK; // 16384 floats
    float* partials = invn + NROWS;               // 512 floats

    k_centroid<<<N_SPK, 256, 0, stream>>>(e, cP);
    k_rownorm<<<NROWS / 4, 128, 0, stream>>>(e, invn);
    k_ge2e<<<NROWS / MROWS, 128, 0, stream>>>(e, cP, invn, w, b, partials);
    k_final<<<1, 256, 0, stream>>>(partials, (float*)d_out);
}